// SGD_NFM_31825707663666
// MI455X (gfx1250) — compile-verified
//
#include <hip/hip_runtime.h>

typedef __attribute__((ext_vector_type(2))) float v2f;
typedef __attribute__((ext_vector_type(8))) float v8f;

#define B_TOT 16384
#define F_SZ  39
#define V_SZ  100000
#define E_SZ  16
#define H_SZ  128

// One block = 16 batch rows (one WMMA M-tile), 256 threads = 8 wave32.
__global__ __launch_bounds__(256) void nfm_fused_kernel(
    const int*   __restrict__ Xi,   // [B,F,1] int32
    const float* __restrict__ Xv,   // [B,F]
    const float* __restrict__ w1,   // [F,V]
    const float* __restrict__ e2,   // [F,V,E]
    const float* __restrict__ W0,   // [E,H]
    const float* __restrict__ b0,   // [H]
    const float* __restrict__ W1,   // [H,H]
    const float* __restrict__ b1,   // [H]
    const float* __restrict__ bb,   // scalar
    float*       __restrict__ out)  // [B]
{
    __shared__ int   sIdx[16 * F_SZ];
    __shared__ float sXv [16 * F_SZ];
    __shared__ float sSec[16][17];     // +1 pad: conflict-free column reads
    __shared__ float sX  [16][132];    // +4 pad: conflict-free column reads
    __shared__ float sFirstP[256];
    __shared__ float sFirst[16];
    __shared__ float sRow[16];

    const int tid   = threadIdx.x;
    const int rowsB = blockIdx.x * 16;      // first batch row of this block

    // Warm W1 (64KB) into cache while we do the gather phase.
    for (int off = tid * 128; off < H_SZ * H_SZ * 4; off += 256 * 128)
        __builtin_prefetch((const char*)W1 + off, 0, 1);

    // Stage indices + values for 16 rows.
    for (int i = tid; i < 16 * F_SZ; i += 256) {
        sIdx[i] = Xi[rowsB * F_SZ + i];
        sXv[i]  = Xv[rowsB * F_SZ + i];
    }
    if (tid < 16) sRow[tid] = 0.0f;
    __syncthreads();

    // ---- Phase 0: embedding gather + FM statistics ----
    {
        const int r = tid >> 4;   // local batch row 0..15
        const int e = tid & 15;   // embedding component
        float s = 0.0f, sq = 0.0f, fo = 0.0f;
        for (int f = 0; f < F_SZ; ++f) {
            const int   idx = sIdx[r * F_SZ + f];
            const float xv  = sXv [r * F_SZ + f];
            const float emb = e2[((size_t)f * V_SZ + (size_t)idx) * E_SZ + e];
            const float val = emb * xv;
            s  += val;
            sq += val * val;
        }
        // first-order term, fields strided across the 16 lanes of this row
        for (int f = e; f < F_SZ; f += 16) {
            const int idx = sIdx[r * F_SZ + f];
            fo += w1[(size_t)f * V_SZ + (size_t)idx] * sXv[r * F_SZ + f];
        }
        sSec[r][e]   = 0.5f * (s * s - sq);   // [16 x 16] A-tile for GEMM1
        sFirstP[tid] = fo;
    }
    __syncthreads();
    if (tid < 16) {
        float acc = 0.0f;
        for (int e = 0; e < 16; ++e) acc += sFirstP[tid * 16 + e];
        sFirst[tid] = acc;
    }
    __syncthreads();

    const int wave = tid >> 5;      // 0..7 -> owns N-tile [wave*16, wave*16+16)
    const int lane = tid & 31;
    const int nl   = lane & 15;     // N within tile / M for A-fragment
    const int kh   = lane >> 4;     // lane-half: K offset selector
    const int n0   = wave * 16;

    // ---- Phase 1: x = relu(second[16x16] @ W0[16x128] + b0) ----
    {
        const float bias = b0[n0 + nl];
        v8f c;
        #pragma unroll
        for (int r = 0; r < 8; ++r) c[r] = bias;   // bias folded into accumulator

        #pragma unroll
        for (int k = 0; k < E_SZ; k += 4) {
            const int ke = k + 2 * kh;
            v2f a, b;
            a.x = sSec[nl][ke];                    // A: M=nl, K=ke
            a.y = sSec[nl][ke + 1];
            b.x = W0[ke * H_SZ + n0 + nl];         // B: K=ke, N=n0+nl
            b.y = W0[(ke + 1) * H_SZ + n0 + nl];
            c = __builtin_amdgcn_wmma_f32_16x16x4_f32(
                    false, a, false, b, (short)0, c, false, false);
        }
        #pragma unroll
        for (int r = 0; r < 8; ++r) {              // C: rows {r, r+8} per half
            float v = c[r];
            v = v > 0.0f ? v : 0.0f;
            sX[r + 8 * kh][n0 + nl] = v;
        }
    }
    __syncthreads();

    // ---- Phase 2: x2 = relu(x[16x128] @ W1[128x128] + b1), row-summed ----
    {
        const float bias = b1[n0 + nl];
        v8f c;
        #pragma unroll
        for (int r = 0; r < 8; ++r) c[r] = bias;

        for (int k = 0; k < H_SZ; k += 4) {
            const int ke = k + 2 * kh;
            v2f a, b;
            a.x = sX[nl][ke];
            a.y = sX[nl][ke + 1];
            b.x = W1[ke * H_SZ + n0 + nl];
            b.y = W1[(ke + 1) * H_SZ + n0 + nl];
            c = __builtin_amdgcn_wmma_f32_16x16x4_f32(
                    false, a, false, b, (short)0, c, false, false);
        }
        #pragma unroll
        for (int r = 0; r < 8; ++r) {
            float v = c[r];
            v = v > 0.0f ? v : 0.0f;
            // reduce over the 16 N-lanes of this half-wave
            v += __shfl_xor(v, 1);
            v += __shfl_xor(v, 2);
            v += __shfl_xor(v, 4);
            v += __shfl_xor(v, 8);
            if (nl == 0) atomicAdd(&sRow[r + 8 * kh], v);   // ds_add_f32
        }
    }
    __syncthreads();

    // ---- Phase 3: final scalar per batch row ----
    if (tid < 16) {
        out[rowsB + tid] = bb[0] + sFirst[tid] + sRow[tid];
    }
}

extern "C" void kernel_launch(void* const* d_in, const int* in_sizes, int n_in,
                              void* d_out, int out_size, void* d_ws, size_t ws_size,
                              hipStream_t stream) {
    const int*   Xi = (const int*)  d_in[0];
    const float* Xv = (const float*)d_in[1];
    const float* w1 = (const float*)d_in[2];
    const float* e2 = (const float*)d_in[3];
    const float* W0 = (const float*)d_in[4];
    const float* b0 = (const float*)d_in[5];
    const float* W1 = (const float*)d_in[6];
    const float* b1 = (const float*)d_in[7];
    const float* bb = (const float*)d_in[8];
    float* out = (float*)d_out;

    const int numBlocks = B_TOT / 16;   // 1024
    nfm_fused_kernel<<<numBlocks, 256, 0, stream>>>(
        Xi, Xv, w1, e2, W0, b0, W1, b1, bb, out);
}